// Sequence_5583457485040
// MI455X (gfx1250) — compile-verified
//
#include <hip/hip_runtime.h>

typedef float v2f __attribute__((ext_vector_type(2)));
typedef float v8f __attribute__((ext_vector_type(8)));

#define DD   128
#define MDIM 512
#define NDIM 512

// ---------------------------------------------------------------------------
// Kernel 1: fp32 WMMA projection (V_WMMA_F32_16X16X4_F32, exact fp32 MAC).
//   part 0: a_out[0][m][g] = x1[m] . W_ih[g][0:128]
//   part 1: a_out[1][n][g] = x2[n] . W_ih[g][128:256] + b_ih[g] + b_hh[g]
// ---------------------------------------------------------------------------
__global__ __launch_bounds__(32, 1)
void proj_wmma(const float* __restrict__ x1,
               const float* __restrict__ x2,
               const float* __restrict__ W,     // (4, 256) row-major
               const float* __restrict__ b_ih,  // (4,)
               const float* __restrict__ b_hh,  // (4,)
               float* __restrict__ a_out)       // [2][512][4]
{
    const int part  = blockIdx.y;             // 0 -> x1/W-left, 1 -> x2/W-right
    const float* X  = part ? x2 : x1;
    const int m0    = blockIdx.x * 16;
    const int lane  = threadIdx.x;
    const int row   = lane & 15;              // A-matrix row (M)
    const int khalf = lane >> 4;              // 0: K=k0,k0+1  1: K=k0+2,k0+3
    const int ncol  = lane & 15;              // B/C column (N)
    const int wcol  = (ncol < 4) ? ncol : 0;  // clamped, in-bounds W row

    const float* xrow = X + (size_t)(m0 + row) * DD;
    const float* wrow = W + (size_t)wcol * (2 * DD) + part * DD;

    v8f acc = {0.f, 0.f, 0.f, 0.f, 0.f, 0.f, 0.f, 0.f};

    #pragma unroll 4
    for (int k0 = 0; k0 < DD; k0 += 4) {
        v2f a;
        a.x = xrow[k0 + khalf * 2 + 0];
        a.y = xrow[k0 + khalf * 2 + 1];

        float w0 = wrow[k0 + khalf * 2 + 0];
        float w1 = wrow[k0 + khalf * 2 + 1];
        v2f b;
        b.x = (ncol < 4) ? w0 : 0.f;
        b.y = (ncol < 4) ? w1 : 0.f;

        // 8-arg form: (neg_a, A, neg_b, B, c_mod, C, reuse_a, reuse_b)
        acc = __builtin_amdgcn_wmma_f32_16x16x4_f32(
                  false, a, false, b, (short)0, acc, false, false);
    }

    float bias = 0.f;
    if (part == 1 && ncol < 4) bias = b_ih[ncol] + b_hh[ncol];

    if (ncol < 4) {
        #pragma unroll
        for (int r = 0; r < 8; ++r) {
            const int mrow = m0 + r + khalf * 8;   // C/D layout: lanes>=16 hold M=r+8
            a_out[part * (MDIM * 4) + mrow * 4 + ncol] = acc[r] + bias;
        }
    }
}

// ---------------------------------------------------------------------------
// Kernel 2: serial scalar LSTM scan (non-associative -> inherently serial).
// One wave; operands pre-scaled by -log2(e) (and -2*log2(e) for the tanh
// gate) so each gate on the h->h critical path is exactly:
//     fma(h, pw_k, base_k) -> v_exp_f32 -> add 1 -> v_rcp_f32
// Chain per step: fma, exp2, add, rcp, fma(c), mul, exp2, add, rcp, fma, mul
// (~11 dependent VALU ops, 2 transcendentals). All per-step operands live in
// LDS. Lanes compute redundantly; lane (n&31) latches h via cndmask and the
// wave issues one coalesced 128B store every 32 steps (off the dep chain).
// ---------------------------------------------------------------------------
#define NL2E  (-1.4426950408889634f)   // -log2(e)
#define NL2E2 (-2.8853900817779268f)   // -2*log2(e)

__global__ __launch_bounds__(32, 1)
void lstm_scan(const float* __restrict__ a_buf,  // [2][512][4] from proj_wmma
               const float* __restrict__ W_hh,   // (4,1)
               float* __restrict__ out)          // (512, 512)
{
    __shared__ float lA1[MDIM * 4];
    __shared__ float lA2[NDIM * 4];

    const int lane = threadIdx.x;

    // cooperative LDS fill, 16B granularity, with per-gate pre-scaling:
    // gates {i,f,o} scaled by -log2(e); gate {g (tanh)} scaled by -2*log2(e)
    const float4* s1 = (const float4*)a_buf;
    const float4* s2 = (const float4*)(a_buf + MDIM * 4);
    float4* d1 = (float4*)lA1;
    float4* d2 = (float4*)lA2;
    for (int i = lane; i < MDIM; i += 32) {
        float4 v1 = s1[i], v2 = s2[i];
        v1.x *= NL2E; v1.y *= NL2E; v1.z *= NL2E2; v1.w *= NL2E;
        v2.x *= NL2E; v2.y *= NL2E; v2.z *= NL2E2; v2.w *= NL2E;
        d1[i] = v1;
        d2[i] = v2;
    }

    const float pw0 = W_hh[0] * NL2E;
    const float pw1 = W_hh[1] * NL2E;
    const float pw2 = W_hh[2] * NL2E2;
    const float pw3 = W_hh[3] * NL2E;
    __syncthreads();

    float h = 0.f, c = 0.f, hold = 0.f;

    for (int m = 0; m < MDIM; ++m) {
        const float a10 = lA1[m * 4 + 0];
        const float a11 = lA1[m * 4 + 1];
        const float a12 = lA1[m * 4 + 2];
        const float a13 = lA1[m * 4 + 3];
        float* orow = out + (size_t)m * NDIM;

        for (int n = 0; n < NDIM; ++n) {
            // off-chain base sums (no dependence on h)
            const float b0 = a10 + lA2[n * 4 + 0];
            const float b1 = a11 + lA2[n * 4 + 1];
            const float b2 = a12 + lA2[n * 4 + 2];
            const float b3 = a13 + lA2[n * 4 + 3];

            // e_k = 2^(-log2e * g_k) = exp(-g_k)   (g gate: exp(-2*g))
            const float e0 = __builtin_amdgcn_exp2f(__builtin_fmaf(h, pw0, b0));
            const float e1 = __builtin_amdgcn_exp2f(__builtin_fmaf(h, pw1, b1));
            const float e2 = __builtin_amdgcn_exp2f(__builtin_fmaf(h, pw2, b2));
            const float e3 = __builtin_amdgcn_exp2f(__builtin_fmaf(h, pw3, b3));

            const float ig = __builtin_amdgcn_rcpf(1.f + e0);  // sigmoid(g0)
            const float fg = __builtin_amdgcn_rcpf(1.f + e1);  // sigmoid(g1)
            const float s2g = __builtin_amdgcn_rcpf(1.f + e2); // sigmoid(2*g2)
            const float og = __builtin_amdgcn_rcpf(1.f + e3);  // sigmoid(g3)
            const float gg = __builtin_fmaf(2.f, s2g, -1.f);   // tanh(g2)

            c = __builtin_fmaf(fg, c, ig * gg);

            // tanh(c) = 2*sigmoid(2c)-1, NaN-free saturation at large |c|
            const float ec = __builtin_amdgcn_exp2f(c * NL2E2);
            const float tc = __builtin_fmaf(2.f, __builtin_amdgcn_rcpf(1.f + ec), -1.f);
            h = og * tc;

            hold = ((n & 31) == lane) ? h : hold;        // latch, cndmask
            if ((n & 31) == 31)                           // uniform branch
                orow[(n & ~31) + lane] = hold;            // coalesced 128B store
        }
    }
}

// ---------------------------------------------------------------------------
extern "C" void kernel_launch(void* const* d_in, const int* in_sizes, int n_in,
                              void* d_out, int out_size, void* d_ws, size_t ws_size,
                              hipStream_t stream) {
    (void)in_sizes; (void)n_in; (void)out_size; (void)ws_size;

    const float* x1   = (const float*)d_in[0];
    const float* x2   = (const float*)d_in[1];
    const float* W_ih = (const float*)d_in[2];
    const float* W_hh = (const float*)d_in[3];
    const float* b_ih = (const float*)d_in[4];
    const float* b_hh = (const float*)d_in[5];
    float* out   = (float*)d_out;
    float* a_buf = (float*)d_ws;   // 2 * 512 * 4 floats = 16 KB scratch

    dim3 grid1(MDIM / 16, 2);
    proj_wmma<<<grid1, 32, 0, stream>>>(x1, x2, W_ih, b_ih, b_hh, a_buf);
    lstm_scan<<<1, 32, 0, stream>>>(a_buf, W_hh, out);
}